// LocalConv2_5059471475010
// MI455X (gfx1250) — compile-verified
//
#include <hip/hip_runtime.h>
#include <math.h>

#define B_   4
#define N_   256
#define C_   32
#define M_   4
#define OUT_ 64

typedef float v2f __attribute__((ext_vector_type(2)));
typedef float v8f __attribute__((ext_vector_type(8)));

// ---------------- workspace layout (float offsets) ----------------
#define OFF_NB   0                               // neighbs      [B,N,4]
#define OFF_RS   (OFF_NB  + B_*N_*M_)            // rowsum/N     [B,N,C]
#define OFF_CS   (OFF_RS  + B_*N_*C_)            // colsum/N     [B,N,C]
#define OFF_DG   (OFF_CS  + B_*N_*C_)            // diag         [B,N,C]
#define OFF_RCS  (OFF_DG  + B_*N_*C_)            // rowcolsum    [B,C]
#define OFF_DS   (OFF_RCS + B_*C_)               // diag mean    [B,C]
#define OFF_S    (OFF_DS  + B_*C_)               // S_m          [B,M,N,C]
#define OFF_T    (OFF_S   + B_*M_*N_*C_)         // T_m          [B,M,N,C]
#define OFF_Q0   (OFF_T   + B_*M_*N_*C_)         // rcs2 core    [B,M,C]
#define OFF_Q1   (OFF_Q0  + B_*M_*C_)            // ds2 core     [B,M,C]
#define OFF_LIN  (OFF_Q1  + B_*M_*C_)            // bias plane   [B,N,OUT]

// ---------------- phase 1: contractions of x ----------------
__global__ void k_contract(const float* __restrict__ x, float* __restrict__ ws) {
  int id = blockIdx.x * 256 + threadIdx.x;          // (b,n,c): B*N*C threads
  int c = id & 31, n = (id >> 5) & 255, b = id >> 13;
  const float* xb = x + (size_t)b * N_ * N_ * C_;
  float srs = 0.f, scs = 0.f;
  for (int i = 0; i < N_; ++i) {
    srs += xb[(i * N_ + n) * C_ + c];               // sum over rows  (axis=1)
    scs += xb[(n * N_ + i) * C_ + c];               // sum over cols  (axis=2)
  }
  ws[OFF_RS + id] = srs * (1.f / N_);
  ws[OFF_CS + id] = scs * (1.f / N_);
  ws[OFF_DG + id] = xb[(n * N_ + n) * C_ + c];
}

__global__ void k_small(float* __restrict__ ws) {
  int id = threadIdx.x;                             // B*C = 128 threads
  int c = id & 31, b = id >> 5;
  float a0 = 0.f, a1 = 0.f;
  for (int n = 0; n < N_; ++n) {
    a0 += ws[OFF_RS + (b * N_ + n) * C_ + c];
    a1 += ws[OFF_DG + (b * N_ + n) * C_ + c];
  }
  ws[OFF_RCS + id] = a0 * (1.f / N_);
  ws[OFF_DS  + id] = a1 * (1.f / N_);
}

// ---------------- phase 1c: neighbs = sigmoid(SnConv2to1) ----------------
__global__ void k_neigh(const float* __restrict__ w1, const float* __restrict__ b1,
                        const float* __restrict__ w0, const float* __restrict__ b0,
                        float* __restrict__ ws) {
  int id = blockIdx.x * 256 + threadIdx.x;          // (b,n): B*N threads
  if (id >= B_ * N_) return;
  int n = id & 255, b = id >> 8;
  float acc[M_];
  #pragma unroll
  for (int m = 0; m < M_; ++m) acc[m] = b1[m] + b0[m];
  for (int c = 0; c < C_; ++c) {
    float r  = ws[OFF_RS  + (b * N_ + n) * C_ + c];
    float cc = ws[OFF_CS  + (b * N_ + n) * C_ + c];
    float d  = ws[OFF_DG  + (b * N_ + n) * C_ + c];
    float rc = ws[OFF_RCS + b * C_ + c];
    float dd = ws[OFF_DS  + b * C_ + c];
    #pragma unroll
    for (int m = 0; m < M_; ++m) {
      acc[m] += r * w1[c * M_ + m] + cc * w1[(C_ + c) * M_ + m] + d * w1[(2 * C_ + c) * M_ + m]
              + rc * w0[c * M_ + m] + dd * w0[(C_ + c) * M_ + m];
    }
  }
  #pragma unroll
  for (int m = 0; m < M_; ++m)
    ws[OFF_NB + id * M_ + m] = 1.f / (1.f + expf(-acc[m]));
}

// ---------------- phase 2a: neighbor-weighted contractions ----------------
__global__ void k_wcontract(const float* __restrict__ x, float* __restrict__ ws) {
  int id = blockIdx.x * 256 + threadIdx.x;          // (b,n,c): B*N*C threads
  int c = id & 31, n = (id >> 5) & 255, b = id >> 13;
  const float* xb = x + (size_t)b * N_ * N_ * C_;
  const float* nb = ws + OFF_NB + b * N_ * M_;
  float s[4] = {0.f, 0.f, 0.f, 0.f}, t[4] = {0.f, 0.f, 0.f, 0.f};
  for (int i = 0; i < N_; ++i) {
    float4 w = *(const float4*)(nb + i * 4);
    float xv = xb[(i * N_ + n) * C_ + c];           // column n, weighted by n_i
    float xw = xb[(n * N_ + i) * C_ + c];           // row n,    weighted by n_j
    s[0] += w.x * xv; s[1] += w.y * xv; s[2] += w.z * xv; s[3] += w.w * xv;
    t[0] += w.x * xw; t[1] += w.y * xw; t[2] += w.z * xw; t[3] += w.w * xw;
  }
  #pragma unroll
  for (int m = 0; m < 4; ++m) {
    ws[OFF_S + ((b * 4 + m) * N_ + n) * C_ + c] = s[m] * (1.f / N_);
    ws[OFF_T + ((b * 4 + m) * N_ + n) * C_ + c] = t[m] * (1.f / N_);
  }
}

__global__ void k_q(float* __restrict__ ws) {
  int id = blockIdx.x * 256 + threadIdx.x;          // (b,m,c): 512 threads
  if (id >= B_ * M_ * C_) return;
  int c = id & 31, m = (id >> 5) & 3, b = id >> 7;
  const float* nb = ws + OFF_NB + b * N_ * M_;
  float a0 = 0.f, a1 = 0.f;
  for (int n = 0; n < N_; ++n) {
    float w = nb[n * 4 + m];
    a0 += w * ws[OFF_S + ((b * 4 + m) * N_ + n) * C_ + c];
    a1 += w * w * ws[OFF_DG + (b * N_ + n) * C_ + c];
  }
  ws[OFF_Q0 + id] = a0 * (1.f / N_);                // rcs2[b, m*C+c]
  ws[OFF_Q1 + id] = a1 * (1.f / N_);                // ds2 [b, m*C+c]
}

// ---------------- phase 2c: fused bias plane linrow[b,j,o] ----------------
__global__ void k_lin(const float* __restrict__ b2d,
                      const float* __restrict__ w1d, const float* __restrict__ b1d,
                      const float* __restrict__ w0d, const float* __restrict__ b0d,
                      float* __restrict__ ws) {
  int id = blockIdx.x * 256 + threadIdx.x;          // (b,j,o): B*N*OUT threads
  int o = id & 63, j = (id >> 6) & 255, b = id >> 14;
  float acc = b2d[o] + b1d[o] + b0d[o];
  #pragma unroll
  for (int m = 0; m < 4; ++m) {
    float nm = ws[OFF_NB + (b * N_ + j) * 4 + m];
    for (int c = 0; c < C_; ++c) {
      int mc = m * C_ + c;
      float Sv = ws[OFF_S + ((b * 4 + m) * N_ + j) * C_ + c];
      float Tv = ws[OFF_T + ((b * 4 + m) * N_ + j) * C_ + c];
      float Dv = ws[OFF_DG + (b * N_ + j) * C_ + c];
      acc += nm * Sv * w1d[mc * OUT_ + o]
           + nm * Tv * w1d[(128 + mc) * OUT_ + o]
           + nm * nm * Dv * w1d[(256 + mc) * OUT_ + o]
           + ws[OFF_Q0 + (b * 4 + m) * C_ + c] * w0d[mc * OUT_ + o]
           + ws[OFF_Q1 + (b * 4 + m) * C_ + c] * w0d[(128 + mc) * OUT_ + o];
    }
  }
  ws[OFF_LIN + id] = acc;
}

// ---------------- phase 3: main WMMA GEMM ----------------
// 1 wave per (b, i, 16-wide j tile). A' rows = n_i,m*n_j,m * x row (half 0)
// and * x-transpose row (half 1); K=256 total vs w2d; OUT=64 -> 4 C-tiles.
__global__ __launch_bounds__(256) void k_main(const float* __restrict__ x,
                                              const float* __restrict__ w2d,
                                              const float* __restrict__ ws,
                                              float* __restrict__ out) {
  __shared__ float lw[16384];                       // 64KB: w2d in B-frag order
  int tid = threadIdx.x;
  #pragma unroll
  for (int p = 0; p < 32; ++p) {                    // 8192 float2 pairs
    int P = p * 256 + tid;
    int l = P & 31, nt = (P >> 5) & 3, ks = (P >> 7) & 31, h = P >> 12;
    int row = h * 128 + ks * 4 + 2 * (l >> 4);
    int col = nt * 16 + (l & 15);
    lw[2 * P]     = w2d[row * OUT_ + col];
    lw[2 * P + 1] = w2d[(row + 1) * OUT_ + col];
  }
  __syncthreads();

  int wid = tid >> 5, lane = tid & 31;
  int tile = blockIdx.x * 8 + wid;                  // 16384 tiles total
  int b   = tile >> 12;
  int rem = tile & 4095;
  int i   = rem >> 4;
  int j0  = (rem & 15) << 4;
  int r = lane & 15, hi = lane >> 4;

  const float* nb = ws + OFF_NB;
  float4 njv = *(const float4*)(nb + (b * N_ + j0 + r) * 4);
  float4 niv = *(const float4*)(nb + (b * N_ + i) * 4);
  float sc[4] = { niv.x * njv.x, niv.y * njv.y, niv.z * njv.z, niv.w * njv.w };

  const float* xa = x + (((size_t)b * N_ + i) * N_ + (j0 + r)) * C_;       // x[b,i,j0+r,:]
  const float* xt = x + (((size_t)b * N_ + (j0 + r)) * N_ + i) * C_;       // x[b,j0+r,i,:]

  v8f acc[4];
  #pragma unroll
  for (int nt = 0; nt < 4; ++nt) acc[nt] = (v8f){0.f,0.f,0.f,0.f,0.f,0.f,0.f,0.f};

  #pragma unroll
  for (int h = 0; h < 2; ++h) {
    const float* xp  = h ? xt : xa;
    const float* lwp = lw + h * 8192;
    #pragma unroll
    for (int ks = 0; ks < 32; ++ks) {
      int m = ks >> 3;
      int c = ((ks & 7) << 2) + 2 * hi;             // lane's K pair within 32-chan slab
      v2f a = *(const v2f*)(xp + c);                // global_load_b64
      float s = sc[m];
      a.x *= s; a.y *= s;
      #pragma unroll
      for (int nt = 0; nt < 4; ++nt) {
        v2f bf = *(const v2f*)(lwp + (ks * 4 + nt) * 64 + lane * 2);  // ds_load_b64
        acc[nt] = __builtin_amdgcn_wmma_f32_16x16x4_f32(
            false, a, false, bf, (short)0, acc[nt], false, false);
      }
    }
  }

  const float* lin = ws + OFF_LIN + (size_t)(b * N_) * OUT_;
  size_t obase = (((size_t)b * N_ + i) * N_) * OUT_;
  #pragma unroll
  for (int nt = 0; nt < 4; ++nt) {
    int o = nt * 16 + r;
    #pragma unroll
    for (int v = 0; v < 8; ++v) {
      int jj = j0 + v + 8 * hi;                     // C-layout: M = v + 8*(lane/16)
      out[obase + (size_t)jj * OUT_ + o] = acc[nt][v] + lin[jj * OUT_ + o];
    }
  }
}

// ---------------- launcher ----------------
extern "C" void kernel_launch(void* const* d_in, const int* in_sizes, int n_in,
                              void* d_out, int out_size, void* d_ws, size_t ws_size,
                              hipStream_t stream) {
  const float* x    = (const float*)d_in[0];
  const float* w1nb = (const float*)d_in[1];
  const float* b1nb = (const float*)d_in[2];
  const float* w0nb = (const float*)d_in[3];
  const float* b0nb = (const float*)d_in[4];
  const float* w2d  = (const float*)d_in[5];
  const float* b2d  = (const float*)d_in[6];
  const float* w1d  = (const float*)d_in[7];
  const float* b1d  = (const float*)d_in[8];
  const float* w0d  = (const float*)d_in[9];
  const float* b0d  = (const float*)d_in[10];
  float* out = (float*)d_out;
  float* ws  = (float*)d_ws;

  k_contract <<<B_ * N_ * C_ / 256, 256, 0, stream>>>(x, ws);
  k_small    <<<1, B_ * C_, 0, stream>>>(ws);
  k_neigh    <<<(B_ * N_ + 255) / 256, 256, 0, stream>>>(w1nb, b1nb, w0nb, b0nb, ws);
  k_wcontract<<<B_ * N_ * C_ / 256, 256, 0, stream>>>(x, ws);
  k_q        <<<2, 256, 0, stream>>>(ws);
  k_lin      <<<B_ * N_ * OUT_ / 256, 256, 0, stream>>>(b2d, w1d, b1d, w0d, b0d, ws);
  k_main     <<<B_ * N_ * (N_ / 16) / 8, 256, 0, stream>>>(x, w2d, ws, out);
}